// RGCN_6468220747930
// MI455X (gfx1250) — compile-verified
//
#include <hip/hip_runtime.h>
#include <hip/hip_bf16.h>

typedef __attribute__((ext_vector_type(2))) float v2f;
typedef __attribute__((ext_vector_type(8))) float v8f;

#define D 512           // D_IN == D_OUT
#define NETYPES 8
#define NNTYPES 4
#define KREL (NETYPES * D)   // 4096
#define MSUB 2               // 2 x 16 = 32 output rows per wave
#define NSUB 4               // 4 x 16 = 64 output columns per wave

// ---------------------------------------------------------------- zero scratch
__global__ __launch_bounds__(256) void rgcn_zero(float* __restrict__ p, long n) {
    long i = (long)blockIdx.x * blockDim.x + threadIdx.x;
    long stride = (long)gridDim.x * blockDim.x;
    for (; i < n; i += stride) p[i] = 0.0f;
}

// ------------------------------------------------- edge scatter: sums + counts
// One wave per edge: lane handles 16 contiguous floats of the 512-wide row.
__global__ __launch_bounds__(256) void rgcn_scatter(
    const float* __restrict__ x_src,
    const int* __restrict__ e_src,
    const int* __restrict__ e_dst,
    const int* __restrict__ e_type,
    float* __restrict__ sums,    // [N_TGT, 8, 512] == [N_TGT, 4096]
    float* __restrict__ counts,  // [N_TGT, 8]
    int E)
{
    int wid  = (int)((blockIdx.x * blockDim.x + threadIdx.x) >> 5);
    int lane = threadIdx.x & 31;
    if (wid >= E) return;
    int src = e_src[wid];
    int dst = e_dst[wid];
    int t   = e_type[wid];

    const float4* xp = (const float4*)(x_src + (size_t)src * D) + (size_t)lane * 4;
    float* sp = sums + ((size_t)dst * NETYPES + t) * D + (size_t)lane * 16;
#pragma unroll
    for (int j = 0; j < 4; ++j) {
        float4 v = xp[j];
        unsafeAtomicAdd(sp + j * 4 + 0, v.x);
        unsafeAtomicAdd(sp + j * 4 + 1, v.y);
        unsafeAtomicAdd(sp + j * 4 + 2, v.z);
        unsafeAtomicAdd(sp + j * 4 + 3, v.w);
    }
    if (lane == 0) unsafeAtomicAdd(&counts[(size_t)dst * NETYPES + t], 1.0f);
}

// ------------------------------------------------------------- 1/max(count,1)
__global__ __launch_bounds__(256) void rgcn_invc(
    const float* __restrict__ counts, float* __restrict__ invc, int n)
{
    int i = blockIdx.x * blockDim.x + threadIdx.x;
    if (i < n) invc[i] = 1.0f / fmaxf(counts[i], 1.0f);
}

// -------------------------------------------------------------- fused WMMA GEMM
// out[m, n] = sum_t mean_A[m, t, :] @ rel_W[t]  +  x_target[m] @ root_W[type(m)]
//            + root_b[type(m)]
// One wave computes a 32 (M) x 64 (N) tile: 2 A-fragments x 4 B-subtiles, so
// each B fragment feeds two v_wmma_f32_16x16x4_f32 ops (0.75 loads / WMMA).
// fp32 A 16x4 layout:  lane L holds row M=L&15; VGPR0 = K=(L>>4)*2, VGPR1 = K+1.
// fp32 B 4x16 layout:  lane L holds col N=L&15; VGPR0 = K=(L>>4)*2, VGPR1 = K+1.
// fp32 C/D 16x16:      VGPR r -> row r + 8*(L>>4), col L&15.
__global__ __launch_bounds__(32) void rgcn_gemm(
    const float* __restrict__ sums,    // [N_TGT, 4096]
    const float* __restrict__ invc,    // [N_TGT, 8]
    const float* __restrict__ x_tgt,   // [N_TGT, 512]
    const float* __restrict__ rel_W,   // [8, 512, 512]
    const float* __restrict__ root_W,  // [4, 512, 512]
    const float* __restrict__ root_b,  // [4, 512]
    const int*   __restrict__ tnt,     // [N_TGT]
    float* __restrict__ out,           // [N_TGT, 512]
    int n_tgt)
{
    const int lane  = threadIdx.x & 31;
    const int mTile = blockIdx.y;                    // 32-row tiles (last one partial)
    const int nBase = blockIdx.x * (16 * NSUB);      // 0..448 step 64
    const int kHalf = (lane >> 4) * 2;               // 0 or 2
    const int nCol  = (lane & 15);

    // Load rows (clamped so the partial last tile reads valid memory; WMMA
    // needs EXEC all-ones, so we clamp loads and guard only the stores).
    const int row0 = mTile * 32 + (lane & 15);
    const int row1 = row0 + 16;
    const int rA0  = row0 < n_tgt ? row0 : n_tgt - 1;
    const int rA1  = row1 < n_tgt ? row1 : n_tgt - 1;

    v8f acc[MSUB][NSUB];
    v8f zero = {};
#pragma unroll
    for (int m = 0; m < MSUB; ++m)
#pragma unroll
        for (int s = 0; s < NSUB; ++s) acc[m][s] = zero;

    // ---- Phase 1: relation GEMM, K = 4096, mean fused via invc scale
    const float* aRow0 = sums + (size_t)rA0 * KREL;
    const float* aRow1 = sums + (size_t)rA1 * KREL;
    for (int t = 0; t < NETYPES; ++t) {
        const float sc0 = invc[(size_t)rA0 * NETYPES + t];
        const float sc1 = invc[(size_t)rA1 * NETYPES + t];
        const float* bBase = rel_W + (size_t)t * D * D;
        const float* aT0 = aRow0 + t * D;
        const float* aT1 = aRow1 + t * D;
        for (int k = 0; k < D; k += 4) {
            const int kr = k + kHalf;
            v2f a0, a1;
            a0.x = aT0[kr] * sc0;  a0.y = aT0[kr + 1] * sc0;
            a1.x = aT1[kr] * sc1;  a1.y = aT1[kr + 1] * sc1;
#pragma unroll
            for (int s = 0; s < NSUB; ++s) {
                v2f b;
                const float* bp = bBase + (size_t)kr * D + nBase + s * 16 + nCol;
                b.x = bp[0];
                b.y = bp[D];
                acc[0][s] = __builtin_amdgcn_wmma_f32_16x16x4_f32(
                    false, a0, false, b, (short)0, acc[0][s], false, false);
                acc[1][s] = __builtin_amdgcn_wmma_f32_16x16x4_f32(
                    false, a1, false, b, (short)0, acc[1][s], false, false);
            }
        }
    }

    // ---- Phase 2: root GEMM, K = 2048, node-type mask fused into A
    const int ty0 = tnt[rA0];
    const int ty1 = tnt[rA1];
    const float* xRow0 = x_tgt + (size_t)rA0 * D;
    const float* xRow1 = x_tgt + (size_t)rA1 * D;
    for (int nt = 0; nt < NNTYPES; ++nt) {
        const float m0 = (ty0 == nt) ? 1.0f : 0.0f;
        const float m1 = (ty1 == nt) ? 1.0f : 0.0f;
        const float* bBase = root_W + (size_t)nt * D * D;
        for (int k = 0; k < D; k += 4) {
            const int kr = k + kHalf;
            v2f a0, a1;
            a0.x = xRow0[kr] * m0;  a0.y = xRow0[kr + 1] * m0;
            a1.x = xRow1[kr] * m1;  a1.y = xRow1[kr + 1] * m1;
#pragma unroll
            for (int s = 0; s < NSUB; ++s) {
                v2f b;
                const float* bp = bBase + (size_t)kr * D + nBase + s * 16 + nCol;
                b.x = bp[0];
                b.y = bp[D];
                acc[0][s] = __builtin_amdgcn_wmma_f32_16x16x4_f32(
                    false, a0, false, b, (short)0, acc[0][s], false, false);
                acc[1][s] = __builtin_amdgcn_wmma_f32_16x16x4_f32(
                    false, a1, false, b, (short)0, acc[1][s], false, false);
            }
        }
    }

    // ---- Epilogue: add root_b[type(row)] and store (guard partial last tile)
#pragma unroll
    for (int ms = 0; ms < MSUB; ++ms) {
#pragma unroll
        for (int s = 0; s < NSUB; ++s) {
            const int col = nBase + s * 16 + nCol;
#pragma unroll
            for (int r = 0; r < 8; ++r) {
                const int row = mTile * 32 + ms * 16 + r + ((lane >> 4) << 3);
                if (row < n_tgt) {
                    const int ty = tnt[row];
                    out[(size_t)row * D + col] = acc[ms][s][r] + root_b[(size_t)ty * D + col];
                }
            }
        }
    }
}

extern "C" void kernel_launch(void* const* d_in, const int* in_sizes, int n_in,
                              void* d_out, int out_size, void* d_ws, size_t ws_size,
                              hipStream_t stream) {
    const float* x_src   = (const float*)d_in[0];
    const float* x_tgt   = (const float*)d_in[1];
    const float* rel_W   = (const float*)d_in[2];
    const float* root_W  = (const float*)d_in[3];
    const float* root_b  = (const float*)d_in[4];
    const int*   e_src   = (const int*)d_in[5];
    const int*   e_dst   = (const int*)d_in[6];
    const int*   e_type  = (const int*)d_in[7];
    const int*   tnt     = (const int*)d_in[8];
    float*       out     = (float*)d_out;

    const int n_tgt = in_sizes[1] / D;     // 10000
    const int E     = in_sizes[5];         // 150000
    const long sums_elems  = (long)n_tgt * KREL;       // 40.96 M floats
    const long count_elems = (long)n_tgt * NETYPES;    // 80 K floats

    float* sums   = (float*)d_ws;
    float* counts = sums + sums_elems;
    float* invc   = counts + count_elems;

    // 1) zero sums + counts (contiguous)
    {
        long n = sums_elems + count_elems;
        int grid = (int)((n + 256L * 8 - 1) / (256L * 8));
        rgcn_zero<<<grid, 256, 0, stream>>>(sums, n);
    }
    // 2) per-edge scatter-add (one wave per edge, 8 edges per 256-thread block)
    {
        int grid = (E + 7) / 8;
        rgcn_scatter<<<grid, 256, 0, stream>>>(x_src, e_src, e_dst, e_type,
                                               sums, counts, E);
    }
    // 3) reciprocal counts
    {
        int n = (int)count_elems;
        rgcn_invc<<<(n + 255) / 256, 256, 0, stream>>>(counts, invc, n);
    }
    // 4) fused WMMA GEMM (rel + root + bias), 32M x 64N per wave
    {
        dim3 grid(D / (16 * NSUB), (n_tgt + 31) / 32);   // (8, 313)
        rgcn_gemm<<<grid, 32, 0, stream>>>(sums, invc, x_tgt, rel_W, root_W,
                                           root_b, tnt, out, n_tgt);
    }
}